// NeighborhoodModel_73504070304089
// MI455X (gfx1250) — compile-verified
//
#include <hip/hip_runtime.h>
#include <math.h>

#define Bb 4
#define Nn 2048
#define Ff 2
#define Ee 16
#define Dd 32
#define Hh 4
#define XDIM (Ee + Ff)
#define BIGNUM 1.0e7f

typedef float v2f __attribute__((ext_vector_type(2)));
typedef float v8f __attribute__((ext_vector_type(8)));

// ---------------- block reduction helpers ----------------
__device__ __forceinline__ float block_reduce_sum(float v, float* sm) {
  int tid = threadIdx.x;
  sm[tid] = v; __syncthreads();
  for (int off = blockDim.x >> 1; off > 0; off >>= 1) {
    if (tid < off) sm[tid] += sm[tid + off];
    __syncthreads();
  }
  float r = sm[0]; __syncthreads();
  return r;
}

__device__ __forceinline__ float block_reduce_max(float v, float* sm) {
  int tid = threadIdx.x;
  sm[tid] = v; __syncthreads();
  for (int off = blockDim.x >> 1; off > 0; off >>= 1) {
    if (tid < off) sm[tid] = fmaxf(sm[tid], sm[tid + off]);
    __syncthreads();
  }
  float r = sm[0]; __syncthreads();
  return r;
}

// ---------------- node encoder MLP ----------------
__global__ void encoder_kernel(const float* __restrict__ feat,
                               const float* __restrict__ emb,
                               const float* __restrict__ We1, const float* __restrict__ be1,
                               const float* __restrict__ We2, const float* __restrict__ be2,
                               float* __restrict__ enc) {
  int idx = blockIdx.x * blockDim.x + threadIdx.x;
  if (idx >= Bb * Nn) return;
  float x[XDIM];
  #pragma unroll
  for (int i = 0; i < Ee; ++i) x[i] = emb[idx * Ee + i];
  x[Ee + 0] = feat[idx * Ff + 0];
  x[Ee + 1] = feat[idx * Ff + 1];
  float h[Dd];
  for (int j = 0; j < Dd; ++j) {
    float s = be1[j];
    for (int i = 0; i < XDIM; ++i) s += x[i] * We1[i * Dd + j];
    h[j] = tanhf(s);
  }
  for (int j = 0; j < Dd; ++j) {
    float s = be2[j];
    for (int i = 0; i < Dd; ++i) s += h[i] * We2[i * Dd + j];
    enc[idx * Dd + j] = tanhf(s);
  }
}

// ---------------- row degrees -> 1/(deg+1) ----------------
__global__ void degree_kernel(const float* __restrict__ adj, float* __restrict__ rowscale) {
  __shared__ float sm[256];
  int row = blockIdx.x;                       // b*Nn + n
  const float* a = adj + (size_t)row * Nn;
  float s = 0.0f;
  for (int m = threadIdx.x; m < Nn; m += blockDim.x) s += a[m];
  float tot = block_reduce_sum(s, sm);
  if (threadIdx.x == 0) rowscale[row] = 1.0f / (tot + 1.0f);
}

// ---------------- WMMA GEMM: agg1 = rowscale .* (adj @ enc) ----------------
// One wave per 16x16 output tile; K loop over 2048 in steps of 4.
__global__ void agg_wmma_kernel(const float* __restrict__ adj,
                                const float* __restrict__ enc,
                                const float* __restrict__ rowscale,
                                float* __restrict__ aggout) {
  int tile = blockIdx.x;          // 0 .. (Nn/16)*(Dd/16)-1
  int b    = blockIdx.y;
  int mt = tile >> 1;             // row tile (Dd/16 == 2 col tiles)
  int nt = tile & 1;              // col tile
  int lane = threadIdx.x;         // 0..31 (wave32)
  const float* A  = adj + (size_t)b * Nn * Nn;
  const float* Bm = enc + (size_t)b * Nn * Dd;

  int arow = mt * 16 + (lane & 15);
  int koff = (lane >> 4) * 2;     // lanes 0-15: K=0,1 ; lanes 16-31: K=2,3
  int ncol = nt * 16 + (lane & 15);

  v8f acc = {0.f, 0.f, 0.f, 0.f, 0.f, 0.f, 0.f, 0.f};
  for (int k = 0; k < Nn; k += 4) {
    const float* ap = A + (size_t)arow * Nn + k + koff;
    v2f av; av.x = ap[0]; av.y = ap[1];                 // A 16x4 layout (ISA 7.12.2)
    v2f bv;                                             // B 4x16 column-striped
    bv.x = Bm[(size_t)(k + koff + 0) * Dd + ncol];
    bv.y = Bm[(size_t)(k + koff + 1) * Dd + ncol];
    acc = __builtin_amdgcn_wmma_f32_16x16x4_f32(
        /*neg_a=*/false, av, /*neg_b=*/false, bv,
        /*c_mod=*/(short)0, acc, /*reuse_a=*/false, /*reuse_b=*/false);
  }
  // C/D layout: VGPR i -> M = i + (lane>>4)*8, N = lane&15
  int mbase = mt * 16 + (lane >> 4) * 8;
  int outn  = nt * 16 + (lane & 15);
  #pragma unroll
  for (int i = 0; i < 8; ++i) {
    int m = mbase + i;
    aggout[((size_t)b * Nn + m) * Dd + outn] = acc[i] * rowscale[b * Nn + m];
  }
}

// ---------------- per-node attention (over L=2 hops) + GRU ----------------
__global__ void layer_post_kernel(const float* __restrict__ enc,
                                  const float* __restrict__ agg1,
                                  const float* __restrict__ w_attn,
                                  const float* __restrict__ Wo, const float* __restrict__ bo,
                                  const float* __restrict__ Wz, const float* __restrict__ Uz, const float* __restrict__ bz,
                                  const float* __restrict__ Wr, const float* __restrict__ Ur, const float* __restrict__ br,
                                  const float* __restrict__ Wh, const float* __restrict__ Uh, const float* __restrict__ bh,
                                  float* __restrict__ encout) {
  int idx = blockIdx.x * blockDim.x + threadIdx.x;
  if (idx >= Bb * Nn) return;
  float e[Dd], a1[Dd], ctx[Dd];
  #pragma unroll
  for (int d = 0; d < Dd; ++d) { e[d] = enc[idx * Dd + d]; a1[d] = agg1[idx * Dd + d]; ctx[d] = 0.0f; }
  // agg0 = enc * 0.5 (identity hop, row-normalized by 1/(1+1))
  for (int h = 0; h < Hh; ++h) {
    float s0 = 0.0f, s1 = 0.0f;
    for (int d = 0; d < Dd; ++d) {
      float w = w_attn[h * Dd + d];
      s0 += tanhf(e[d] * 0.5f) * w;
      s1 += tanhf(a1[d]) * w;
    }
    float mx = fmaxf(s0, s1);
    float e0 = expf(s0 - mx), e1 = expf(s1 - mx);
    float inv = 1.0f / (e0 + e1);
    float al0 = e0 * inv, al1 = e1 * inv;
    for (int d = 0; d < Dd; ++d) ctx[d] += al0 * (e[d] * 0.5f) + al1 * a1[d];
  }
  for (int d = 0; d < Dd; ++d) ctx[d] *= (1.0f / Hh);
  float nbr[Dd];
  for (int j = 0; j < Dd; ++j) {
    float s = bo[j];
    for (int d = 0; d < Dd; ++d) s += ctx[d] * Wo[d * Dd + j];
    nbr[j] = tanhf(s);
  }
  float re[Dd];
  for (int j = 0; j < Dd; ++j) {
    float s = br[j];
    for (int d = 0; d < Dd; ++d) s += nbr[d] * Wr[d * Dd + j] + e[d] * Ur[d * Dd + j];
    re[j] = (1.0f / (1.0f + expf(-s))) * e[j];
  }
  for (int j = 0; j < Dd; ++j) {
    float sh = bh[j], sz = bz[j];
    for (int d = 0; d < Dd; ++d) {
      sh += nbr[d] * Wh[d * Dd + j] + re[d] * Uh[d * Dd + j];
      sz += nbr[d] * Wz[d * Dd + j] + e[d] * Uz[d * Dd + j];
    }
    float hv = tanhf(sh);
    float zv = 1.0f / (1.0f + expf(-sz));
    encout[idx * Dd + j] = zv * e[j] + (1.0f - zv) * hv;
  }
}

// ---------------- primal/dual heads + s0 ----------------
__global__ void head_kernel(const float* __restrict__ enc,
                            const float* __restrict__ demands,
                            const float* __restrict__ Wd1, const float* __restrict__ bd1,
                            const float* __restrict__ Wd2, const float* __restrict__ bd2,
                            const float* __restrict__ Wv1, const float* __restrict__ bv1,
                            const float* __restrict__ Wv2, const float* __restrict__ bv2,
                            float* __restrict__ pred, float* __restrict__ dualv,
                            float* __restrict__ s0) {
  int idx = blockIdx.x * blockDim.x + threadIdx.x;
  if (idx >= Bb * Nn) return;
  float e[Dd];
  #pragma unroll
  for (int d = 0; d < Dd; ++d) e[d] = enc[idx * Dd + d];
  float p = bd2[0];
  for (int j = 0; j < Dd; ++j) {
    float s = bd1[j];
    for (int d = 0; d < Dd; ++d) s += e[d] * Wd1[d * Dd + j];
    p += s * Wd2[j];
  }
  pred[idx] = p;
  float v = bv2[0];
  for (int j = 0; j < Dd; ++j) {
    float s = bv1[j];
    for (int d = 0; d < Dd; ++d) s += e[d] * Wv1[d * Dd + j];
    v += s * Wv2[j];
  }
  dualv[idx] = v;
  s0[idx] = fmaxf(-demands[idx], 0.0f);
}

// ---------------- masked-softmax row stats: max, 1/sum, sum(fwp^2) ----------------
__global__ void rowstats_kernel(const float* __restrict__ adj,
                                const float* __restrict__ pred,
                                float* __restrict__ rmax, float* __restrict__ irs,
                                float* __restrict__ rowsq) {
  __shared__ float sm[256];
  int row = blockIdx.x;                 // b*Nn + n
  int b = row / Nn;
  const float* a = adj + (size_t)row * Nn;
  const float* pm = pred + b * Nn;
  float pn = pred[row];
  float mx = -3.0e38f;
  for (int m = threadIdx.x; m < Nn; m += blockDim.x) {
    float av = a[m];
    float w = -BIGNUM * (1.0f - av) + pn * pm[m];
    mx = fmaxf(mx, w);
  }
  float bmax = block_reduce_max(mx, sm);
  float s = 0.0f, s2 = 0.0f;
  for (int m = threadIdx.x; m < Nn; m += blockDim.x) {
    float av = a[m];
    float w = -BIGNUM * (1.0f - av) + pn * pm[m];
    float ev = expf(w - bmax);
    s += ev; s2 += ev * ev;
  }
  float S = block_reduce_sum(s, sm);
  float S2 = block_reduce_sum(s2, sm);
  if (threadIdx.x == 0) {
    rmax[row] = bmax;
    irs[row] = 1.0f / S;
    rowsq[row] = S2 / (S * S);
  }
}

// ---------------- one flow iteration: s_out[m] = relu( (fwp^T s_in)[m] - d[m] ) ----------------
__global__ void flow_iter_kernel(const float* __restrict__ adj,
                                 const float* __restrict__ pred,
                                 const float* __restrict__ rmax,
                                 const float* __restrict__ irs,
                                 const float* __restrict__ demands,
                                 const float* __restrict__ sin_,
                                 float* __restrict__ sout) {
  int b = blockIdx.y;
  int m = blockIdx.x * blockDim.x + threadIdx.x;
  const float* acol = adj + (size_t)b * Nn * Nn + m;   // column m, stride Nn
  float pmv = pred[b * Nn + m];
  float t = 0.0f;
  for (int n = 0; n < Nn; ++n) {
    float av = acol[(size_t)n * Nn];
    if (av != 0.0f) {                                   // masked entries: expf(-1e7)==0 exactly
      int rn = b * Nn + n;
      t += av * expf(pred[rn] * pmv - rmax[rn]) * irs[rn] * sin_[rn];
    }
  }
  int idx = b * Nn + m;
  sout[idx] = fmaxf(t - demands[idx], 0.0f);
}

// ---------------- dual cost quadratic part: sum over edges of df^2 - dd*df ----------------
__global__ void dual_row_kernel(const float* __restrict__ adj,
                                const float* __restrict__ dualv,
                                float* __restrict__ acc) {
  __shared__ float sm[256];
  int n = blockIdx.x;
  int b = blockIdx.y;
  const float* a = adj + ((size_t)b * Nn + n) * Nn;
  float dn = dualv[b * Nn + n];
  float local = 0.0f;
  for (int m = threadIdx.x; m < Nn; m += blockDim.x) {
    float av = a[m];
    if (av != 0.0f) {
      float dd = dn - dualv[b * Nn + m];
      float f = av * fmaxf(dd * 0.5f, 0.0f);
      local += f * f - dd * f;
    }
  }
  float tot = block_reduce_sum(local, sm);
  if (threadIdx.x == 0) atomicAdd(&acc[b], tot);
}

// ---------------- finalize: out[b] = flow_cost - dual_cost ----------------
__global__ void finalize_kernel(const float* __restrict__ s10,
                                const float* __restrict__ rowsq,
                                const float* __restrict__ dualv,
                                const float* __restrict__ demands,
                                const float* __restrict__ dualacc,
                                float* __restrict__ out) {
  __shared__ float sm[256];
  int b = blockIdx.x;
  float fc = 0.0f, dvd = 0.0f;
  for (int n = threadIdx.x; n < Nn; n += blockDim.x) {
    int idx = b * Nn + n;
    float sv = s10[idx];
    fc += sv * sv * rowsq[idx];
    dvd += dualv[idx] * demands[idx];
  }
  float FC = block_reduce_sum(fc, sm);
  float DVD = block_reduce_sum(dvd, sm);
  if (threadIdx.x == 0) {
    float dual_cost = dualacc[b] - DVD;
    out[b] = FC - dual_cost;
  }
}

extern "C" void kernel_launch(void* const* d_in, const int* in_sizes, int n_in,
                              void* d_out, int out_size, void* d_ws, size_t ws_size,
                              hipStream_t stream) {
  (void)in_sizes; (void)n_in; (void)out_size; (void)ws_size;
  const float* feat    = (const float*)d_in[0];
  const float* emb     = (const float*)d_in[1];
  const float* demands = (const float*)d_in[2];
  const float* adj     = (const float*)d_in[3];
  // d_in[4] = neighborhoods: unused (hop0 = I/2 analytically, hop1 = adj/(deg+1))
  const float* We1 = (const float*)d_in[5];  const float* be1 = (const float*)d_in[6];
  const float* We2 = (const float*)d_in[7];  const float* be2 = (const float*)d_in[8];
  const float* w_attn = (const float*)d_in[9];
  const float* Wo = (const float*)d_in[10];  const float* bo = (const float*)d_in[11];
  const float* Wz = (const float*)d_in[12];  const float* Uz = (const float*)d_in[13]; const float* bz = (const float*)d_in[14];
  const float* Wr = (const float*)d_in[15];  const float* Ur = (const float*)d_in[16]; const float* br = (const float*)d_in[17];
  const float* Wh = (const float*)d_in[18];  const float* Uh = (const float*)d_in[19]; const float* bh = (const float*)d_in[20];
  const float* Wd1 = (const float*)d_in[21]; const float* bd1 = (const float*)d_in[22];
  const float* Wd2 = (const float*)d_in[23]; const float* bd2 = (const float*)d_in[24];
  const float* Wv1 = (const float*)d_in[25]; const float* bv1 = (const float*)d_in[26];
  const float* Wv2 = (const float*)d_in[27]; const float* bv2 = (const float*)d_in[28];

  float* W = (float*)d_ws;
  float* E0       = W;                         // Bb*Nn*Dd
  float* E1       = E0 + (size_t)Bb * Nn * Dd;
  float* AGG      = E1 + (size_t)Bb * Nn * Dd;
  float* ROWSCALE = AGG + (size_t)Bb * Nn * Dd; // Bb*Nn
  float* PRED     = ROWSCALE + Bb * Nn;
  float* DUALV    = PRED + Bb * Nn;
  float* RMAX     = DUALV + Bb * Nn;
  float* IRS      = RMAX + Bb * Nn;
  float* RSQ      = IRS + Bb * Nn;
  float* S0       = RSQ + Bb * Nn;
  float* S1       = S0 + Bb * Nn;
  float* ACC      = S1 + Bb * Nn;              // Bb

  int nodeBlocks = (Bb * Nn + 255) / 256;

  encoder_kernel<<<nodeBlocks, 256, 0, stream>>>(feat, emb, We1, be1, We2, be2, E0);
  degree_kernel<<<Bb * Nn, 256, 0, stream>>>(adj, ROWSCALE);

  for (int layer = 0; layer < 3; ++layer) {
    const float* ein = (layer & 1) ? E1 : E0;
    float* eout      = (layer & 1) ? E0 : E1;
    agg_wmma_kernel<<<dim3((Nn / 16) * (Dd / 16), Bb), 32, 0, stream>>>(adj, ein, ROWSCALE, AGG);
    layer_post_kernel<<<nodeBlocks, 256, 0, stream>>>(ein, AGG, w_attn, Wo, bo,
                                                      Wz, Uz, bz, Wr, Ur, br, Wh, Uh, bh, eout);
  }
  const float* encF = E1;  // after 3 layers: E0->E1, E1->E0, E0->E1

  head_kernel<<<nodeBlocks, 256, 0, stream>>>(encF, demands, Wd1, bd1, Wd2, bd2,
                                              Wv1, bv1, Wv2, bv2, PRED, DUALV, S0);
  rowstats_kernel<<<Bb * Nn, 256, 0, stream>>>(adj, PRED, RMAX, IRS, RSQ);

  for (int it = 0; it < 10; ++it) {
    const float* si = (it & 1) ? S1 : S0;
    float* so       = (it & 1) ? S0 : S1;
    flow_iter_kernel<<<dim3(Nn / 256, Bb), 256, 0, stream>>>(adj, PRED, RMAX, IRS, demands, si, so);
  }
  // 10 iterations: final s_10 lands in S0

  hipMemsetAsync(ACC, 0, Bb * sizeof(float), stream);
  dual_row_kernel<<<dim3(Nn, Bb), 256, 0, stream>>>(adj, DUALV, ACC);
  finalize_kernel<<<Bb, 256, 0, stream>>>(S0, RSQ, DUALV, demands, ACC, (float*)d_out);
}